// LineLayer_42253888258092
// MI455X (gfx1250) — compile-verified
//
#include <hip/hip_runtime.h>
#include <hip/hip_bf16.h>
#include <math.h>

// ---------------- problem constants ----------------
#define NN   50000
#define EE   300000
#define ELL  600000
#define DD   256
#define BB   512

// flat d_out offsets (floats), in reference return order
#define OFF_ATOM   0L
#define OFF_POS    76800000L                 // E*D
#define OFF_BATCHE 76950000L                 // + N*3
#define OFF_MOL    77250000L                 // + E
#define OFF_LGEI   77381072L                 // + B*D
#define OFF_NEA    78581072L                 // + 2*EL

// workspace layout (floats)
#define WS_XLIN 0L                            // N*D
#define WS_CAT  ((long)NN * DD)               // B*512 : [b][0:256]=sum acc, [b][256:512]=max acc

typedef __attribute__((ext_vector_type(2))) float v2f;
typedef __attribute__((ext_vector_type(4))) float v4f;
typedef __attribute__((ext_vector_type(8))) float v8f;

// ---------------------------------------------------------------------------
// WMMA fp32 GEMM:  out[M,N] = A[M,K] @ W[N,K]^T + bias[N]
// one wave per 16x16 tile, K stepped by 4 via V_WMMA_F32_16X16X4_F32.
// fp32 A 16x4 layout: lane = 16*khi + m, regs = A[m][kk+2*khi .. +1];
// B[k][n] = W[n][k] gives the same contiguous float2 pattern on the W side.
// ---------------------------------------------------------------------------
__global__ __launch_bounds__(32) void wmma_gemm_f32(
    const float* __restrict__ A, const float* __restrict__ W,
    const float* __restrict__ bias, float* __restrict__ out,
    int M, int N, int K)
{
    const int tm   = blockIdx.x;        // tile row
    const int tn   = blockIdx.y;        // tile col
    const int lane = threadIdx.x;       // wave32
    const int m    = lane & 15;         // also the n index for the B fragment
    const int khi  = lane >> 4;

    const float* arow = A + (size_t)(tm * 16 + m) * K + 2 * khi;
    const float* wrow = W + (size_t)(tn * 16 + m) * K + 2 * khi;

    v8f c = {};
    #pragma unroll 4
    for (int kk = 0; kk < K; kk += 4) {
        v2f a = *(const v2f*)(arow + kk);
        v2f b = *(const v2f*)(wrow + kk);
        // (neg_a, A, neg_b, B, c_mod, C, reuse_a, reuse_b)
        c = __builtin_amdgcn_wmma_f32_16x16x4_f32(false, a, false, b,
                                                  (short)0, c, false, false);
    }

    const float bv = bias[tn * 16 + m];
    float* obase = out + (size_t)(tm * 16) * N + tn * 16 + m;
    #pragma unroll
    for (int r = 0; r < 8; ++r) {
        int row = r + 8 * khi;                 // C/D layout: M = r + 8*khi, N = lane&15
        obase[(size_t)row * N] = c[r] + bv;
    }
}

// ---------------------------------------------------------------------------
// init the segment accumulator buffer: [b][0:256]=0 (sum), [b][256:512]=-inf (max)
// ---------------------------------------------------------------------------
__global__ void init_cat_kernel(float* __restrict__ cat)
{
    int i = blockIdx.x * 256 + threadIdx.x;    // B*512 = 262144 total
    cat[i] = ((i & 511) < 256) ? 0.0f : -INFINITY;
}

// ---------------------------------------------------------------------------
// per-edge pipeline: h=elu(xl[u]+xl[v]); atom=elu(h*attn);
// score=sigmoid(atom.rw + rb); segment sum/max into cat; emit atom + batch_e.
// one 256-thread block per edge; x_lin gathers hit L2 (51MB table).
// ---------------------------------------------------------------------------
__global__ __launch_bounds__(256) void edge_kernel(
    const float* __restrict__ xlin, const int* __restrict__ edges,
    const int* __restrict__ batch, const float* __restrict__ attn,
    const float* __restrict__ rw,  const float* __restrict__ rb,
    float* __restrict__ out_atom, float* __restrict__ out_batch_e,
    float* __restrict__ cat)
{
    __shared__ float red[256];
    const int e = blockIdx.x;
    const int t = threadIdx.x;

    const int u = edges[2 * e + 0];
    const int v = edges[2 * e + 1];

    float s = xlin[(size_t)u * DD + t] + xlin[(size_t)v * DD + t];
    float h = (s > 0.0f) ? s : (expf(s) - 1.0f);
    float ha = h * attn[t];
    float atom = (ha > 0.0f) ? ha : (expf(ha) - 1.0f);

    red[t] = atom * rw[t];
    __syncthreads();
    #pragma unroll
    for (int w = 128; w > 0; w >>= 1) {
        if (t < w) red[t] += red[t + w];
        __syncthreads();
    }
    float score = 1.0f / (1.0f + expf(-(red[0] + rb[0])));

    // streaming 307MB output: non-temporal so x_lin stays L2-resident
    __builtin_nontemporal_store(atom, &out_atom[(size_t)e * DD + t]);

    const int b = batch[u];
    if (t == 0) out_batch_e[e] = (float)b;

    atomicAdd(&cat[(size_t)b * 512 + t], score * atom);        // segment_sum part
    atomicMax(&cat[(size_t)b * 512 + 256 + t], atom);          // segment_max part (f32)
}

// ---------------------------------------------------------------------------
// new_edge_attr: (x[ea0] + x[shared] + x[node_b]) / 3  (EL x 256)
// 4 rows per 256-thread block, float4 lanes; NT stores (614MB stream).
// ---------------------------------------------------------------------------
__global__ __launch_bounds__(256) void line_graph_kernel(
    const float* __restrict__ x, const int* __restrict__ edge_idx,
    const int* __restrict__ lgei, float* __restrict__ out_nea)
{
    long row = (long)blockIdx.x * 4 + (threadIdx.x >> 6);
    int  t   = threadIdx.x & 63;                // 64 float4 per 256-float row
    if (row >= ELL) return;

    int l0 = lgei[row];             // line_graph_edge_index[0][row]
    int l1 = lgei[ELL + row];       // line_graph_edge_index[1][row]

    int ea0    = edge_idx[l0];           // edge_idx[0][l0]
    int shared = edge_idx[EE + l0];      // edge_idx[1][l0]
    int eb0    = edge_idx[l1];           // edge_idx[0][l1]
    int eb1    = edge_idx[EE + l1];      // edge_idx[1][l1]
    int node_b = (shared == eb1) ? eb1 : eb0;

    const v4f* xr = (const v4f*)x;       // D=256 -> 64 v4f per row
    v4f a = xr[(size_t)ea0    * 64 + t];
    v4f b = xr[(size_t)shared * 64 + t];
    v4f c = xr[(size_t)node_b * 64 + t];
    v4f r = (a + b + c) * (1.0f / 3.0f);

    __builtin_nontemporal_store(r, ((v4f*)out_nea) + row * 64 + t);
}

// ---------------------------------------------------------------------------
// int -> float passthrough for line_graph_edge_index output
// ---------------------------------------------------------------------------
__global__ void lgei_cast_kernel(const int* __restrict__ lgei, float* __restrict__ out)
{
    long i = (long)blockIdx.x * 256 + threadIdx.x;
    if (i < 2L * ELL) out[i] = (float)lgei[i];
}

// ---------------------------------------------------------------------------
extern "C" void kernel_launch(void* const* d_in, const int* in_sizes, int n_in,
                              void* d_out, int out_size, void* d_ws, size_t ws_size,
                              hipStream_t stream)
{
    const float* x        = (const float*)d_in[0];
    const int*   edges    = (const int*)  d_in[1];
    const float* pos      = (const float*)d_in[2];
    const int*   batch    = (const int*)  d_in[3];
    const int*   edge_idx = (const int*)  d_in[4];
    // d_in[5] = edge_attr (unused by the reference outputs)
    const int*   lgei     = (const int*)  d_in[6];
    const float* linear_w = (const float*)d_in[7];
    const float* linear_b = (const float*)d_in[8];
    const float* attn     = (const float*)d_in[9];
    const float* rw       = (const float*)d_in[10];
    const float* rb       = (const float*)d_in[11];
    const float* lin2_w   = (const float*)d_in[12];
    const float* lin2_b   = (const float*)d_in[13];

    float* out  = (float*)d_out;
    float* xlin = (float*)d_ws + WS_XLIN;
    float* cat  = (float*)d_ws + WS_CAT;

    // 1) reset segment accumulators (sum=0 / max=-inf) -- every call, deterministic
    init_cat_kernel<<<(BB * 512) / 256, 256, 0, stream>>>(cat);

    // 2) x_lin = x @ linear_w^T + linear_b   (WMMA fp32, 3125x16 tiles)
    {
        dim3 grid(NN / 16, DD / 16);
        wmma_gemm_f32<<<grid, 32, 0, stream>>>(x, linear_w, linear_b, xlin,
                                               NN, DD, DD);
    }

    // 3) per-edge pipeline + segment sum/max into cat + atom_repr + batch_e
    edge_kernel<<<EE, 256, 0, stream>>>(xlin, edges, batch, attn, rw, rb,
                                        out + OFF_ATOM, out + OFF_BATCHE, cat);

    // 4) mol_repr = cat @ lin2_w^T + lin2_b   (WMMA fp32, 32x16 tiles, K=512)
    {
        dim3 grid(BB / 16, DD / 16);
        wmma_gemm_f32<<<grid, 32, 0, stream>>>(cat, lin2_w, lin2_b, out + OFF_MOL,
                                               BB, DD, 2 * DD);
    }

    // 5) new_edge_attr (independent of 1-4)
    line_graph_kernel<<<ELL / 4, 256, 0, stream>>>(x, edge_idx, lgei, out + OFF_NEA);

    // 6) passthrough outputs
    hipMemcpyAsync(out + OFF_POS, pos, (size_t)NN * 3 * sizeof(float),
                   hipMemcpyDeviceToDevice, stream);
    lgei_cast_kernel<<<(2 * ELL + 255) / 256, 256, 0, stream>>>(lgei, out + OFF_LGEI);
}